// NMSLayer_6210522710543
// MI455X (gfx1250) — compile-verified
//
#include <hip/hip_runtime.h>
#include <stdint.h>

typedef unsigned int u32;
typedef unsigned int uv4 __attribute__((ext_vector_type(4)));
typedef int          iv8 __attribute__((ext_vector_type(8)));
typedef int          iv4 __attribute__((ext_vector_type(4)));

#define HW    2048
#define TILE  64
#define HALO  3
#define IN_DIM (TILE + 2 * HALO)   // 70
#define NBINS 65536

// ---------- monotone float -> u32 key (order-preserving) ----------
__device__ __forceinline__ u32 fkey(u32 u) {
    u32 mask = (u32)(((int)u) >> 31) | 0x80000000u;
    return u ^ mask;
}

// ---------- workspace zeroing ----------
__global__ __launch_bounds__(256) void zero_kernel(u32* ws, int cnt) {
    int i = blockIdx.x * blockDim.x + threadIdx.x;
    if (i < cnt) ws[i] = 0u;
}

// ---------- pass 1 (LDS-privatized): per-WG 64Ki-bin histogram in 256KB LDS ----------
__global__ __launch_bounds__(1024) void hist_hi_lds_kernel(const float* __restrict__ x, int n,
                                                           u32* __restrict__ copies) {
    extern __shared__ u32 lh[];                       // 65536 bins = 256 KB (CDNA5: 320KB/WGP)
    for (int i = threadIdx.x; i < NBINS; i += blockDim.x) lh[i] = 0u;
    __syncthreads();

    int stride = gridDim.x * blockDim.x;
    int gtid = blockIdx.x * blockDim.x + threadIdx.x;
    int n4 = n >> 2;
    const float4* x4 = (const float4*)x;
    for (int i = gtid; i < n4; i += stride) {
        float4 v = x4[i];
        atomicAdd(&lh[fkey(__float_as_uint(v.x)) >> 16], 1u);
        atomicAdd(&lh[fkey(__float_as_uint(v.y)) >> 16], 1u);
        atomicAdd(&lh[fkey(__float_as_uint(v.z)) >> 16], 1u);
        atomicAdd(&lh[fkey(__float_as_uint(v.w)) >> 16], 1u);
    }
    for (int i = (n4 << 2) + gtid; i < n; i += stride)
        atomicAdd(&lh[fkey(__float_as_uint(x[i])) >> 16], 1u);
    __syncthreads();

    u32* dst = copies + (size_t)blockIdx.x * NBINS;   // plain stores, no atomics
    for (int i = threadIdx.x; i < NBINS; i += blockDim.x) dst[i] = lh[i];
}

// ---------- contention-free merge of per-WG histogram copies ----------
__global__ __launch_bounds__(256) void reduce_hist_kernel(const u32* __restrict__ copies,
                                                          u32* __restrict__ hist, int ncopies) {
    int bin = blockIdx.x * blockDim.x + threadIdx.x;  // 65536 total
    u32 s = 0;
    for (int c = 0; c < ncopies; ++c) s += copies[(size_t)c * NBINS + bin];
    hist[bin] = s;
}

// ---------- pass 1 fallback (global atomics) when scratch is tight ----------
__global__ __launch_bounds__(256) void hist_hi_kernel(const float* __restrict__ x, int n,
                                                      u32* __restrict__ hist) {
    int stride = gridDim.x * blockDim.x;
    int gtid = blockIdx.x * blockDim.x + threadIdx.x;
    int n4 = n >> 2;
    const float4* x4 = (const float4*)x;
    for (int i = gtid; i < n4; i += stride) {
        float4 v = x4[i];
        atomicAdd(&hist[fkey(__float_as_uint(v.x)) >> 16], 1u);
        atomicAdd(&hist[fkey(__float_as_uint(v.y)) >> 16], 1u);
        atomicAdd(&hist[fkey(__float_as_uint(v.z)) >> 16], 1u);
        atomicAdd(&hist[fkey(__float_as_uint(v.w)) >> 16], 1u);
    }
    for (int i = (n4 << 2) + gtid; i < n; i += stride)
        atomicAdd(&hist[fkey(__float_as_uint(x[i])) >> 16], 1u);
}

// ---------- select the bucket containing rank k (single 256-thread WG) ----------
__global__ __launch_bounds__(256) void select_hi_kernel(const u32* __restrict__ hist,
                                                        u32* __restrict__ scalars, u32 k) {
    __shared__ u32 part[256];
    int t = threadIdx.x;
    u32 sum = 0;
    for (int i = 0; i < 256; ++i) sum += hist[(t << 8) + i];
    part[t] = sum;
    __syncthreads();
    u32 incl = sum;
    for (int off = 1; off < 256; off <<= 1) {
        u32 v = (t >= off) ? part[t - off] : 0u;
        __syncthreads();
        incl += v;
        part[t] = incl;
        __syncthreads();
    }
    u32 excl = incl - sum;
    if (excl <= k && k < incl) {
        u32 cum = excl;
        for (int i = 0; i < 256; ++i) {
            u32 h = hist[(t << 8) + i];
            if (k < cum + h) { scalars[0] = (u32)((t << 8) + i); scalars[1] = cum; break; }
            cum += h;
        }
    }
}

// ---------- pass 2: low 16 bits within winning bucket (few hits -> global atomics OK) ----------
__global__ __launch_bounds__(256) void hist_lo_kernel(const float* __restrict__ x, int n,
                                                      const u32* __restrict__ scalars,
                                                      u32* __restrict__ hist2) {
    u32 B = scalars[0];
    int stride = gridDim.x * blockDim.x;
    int gtid = blockIdx.x * blockDim.x + threadIdx.x;
    int n4 = n >> 2;
    const float4* x4 = (const float4*)x;
    for (int i = gtid; i < n4; i += stride) {
        float4 v = x4[i];
        u32 k0 = fkey(__float_as_uint(v.x));
        u32 k1 = fkey(__float_as_uint(v.y));
        u32 k2 = fkey(__float_as_uint(v.z));
        u32 k3 = fkey(__float_as_uint(v.w));
        if ((k0 >> 16) == B) atomicAdd(&hist2[k0 & 0xFFFFu], 1u);
        if ((k1 >> 16) == B) atomicAdd(&hist2[k1 & 0xFFFFu], 1u);
        if ((k2 >> 16) == B) atomicAdd(&hist2[k2 & 0xFFFFu], 1u);
        if ((k3 >> 16) == B) atomicAdd(&hist2[k3 & 0xFFFFu], 1u);
    }
    for (int i = (n4 << 2) + gtid; i < n; i += stride) {
        u32 kk = fkey(__float_as_uint(x[i]));
        if ((kk >> 16) == B) atomicAdd(&hist2[kk & 0xFFFFu], 1u);
    }
}

// ---------- final select: produce median float bits in scalars[2] ----------
__global__ __launch_bounds__(256) void select_lo_kernel(const u32* __restrict__ hist2,
                                                        u32* __restrict__ scalars, u32 k) {
    __shared__ u32 part[256];
    int t = threadIdx.x;
    u32 base = scalars[1];
    u32 B = scalars[0];
    u32 sum = 0;
    for (int i = 0; i < 256; ++i) sum += hist2[(t << 8) + i];
    part[t] = sum;
    __syncthreads();
    u32 incl = sum;
    for (int off = 1; off < 256; off <<= 1) {
        u32 v = (t >= off) ? part[t - off] : 0u;
        __syncthreads();
        incl += v;
        part[t] = incl;
        __syncthreads();
    }
    u32 excl = incl - sum;
    if (base + excl <= k && k < base + incl) {
        u32 cum = base + excl;
        for (int i = 0; i < 256; ++i) {
            u32 h = hist2[(t << 8) + i];
            if (k < cum + h) {
                u32 key = (B << 16) | (u32)((t << 8) + i);
                u32 u = (key & 0x80000000u) ? (key ^ 0x80000000u) : ~key;
                scalars[2] = u;   // median as float bits
                break;
            }
            cum += h;
        }
    }
}

// ---------- fused threshold + 7x7 maxpool + NMS mask (TDM tile load) ----------
__global__ __launch_bounds__(256) void nms_kernel(const float* __restrict__ x,
                                                  const u32* __restrict__ scalars,
                                                  float* __restrict__ out) {
    __shared__ float raw[IN_DIM * IN_DIM];        // TDM target, row stride = cols
    __shared__ float hmax[IN_DIM * (TILE + 1)];   // horizontal-max stage

    const int tx = blockIdx.x, ty = blockIdx.y, p = blockIdx.z;
    const int row0 = ty * TILE, col0 = tx * TILE;
    const int r_lo = max(row0 - HALO, 0);
    const int r_hi = min(row0 + TILE + HALO, HW);
    const int c_lo = max(col0 - HALO, 0);
    const int c_hi = min(col0 + TILE + HALO, HW);
    const int rows = r_hi - r_lo;
    const int cols = c_hi - c_lo;

    const float med = __uint_as_float(scalars[2]);
    const float NEG_INF = -__builtin_inff();
    const float* plane = x + (size_t)p * HW * HW;

#if __has_builtin(__builtin_amdgcn_tensor_load_to_lds)
    // One TDM descriptor per tile, issued by wave 0 only (wave32).
    if (threadIdx.x < 32u) {
        uint64_t ga = (uint64_t)(uintptr_t)(plane + (size_t)r_lo * HW + c_lo);
        u32 lds_addr = (u32)(uintptr_t)(&raw[0]);   // low 32 bits of LDS-aperture addr
        uv4 g0 = { 1u,                               // count=1, user desc
                   lds_addr,
                   (u32)ga,
                   (u32)((ga >> 32) & 0x01FFFFFFu) | 0x80000000u };  // type=2
        iv8 g1 = { (int)(2u << 16),                                  // data_size = 4B
                   (int)((HW & 0xFFFF) << 16),                       // tensor_dim0 lo16
                   (int)(((HW >> 16) & 0xFFFF) | ((HW & 0xFFFF) << 16)), // dim0 hi | dim1 lo
                   (int)(((HW >> 16) & 0xFFFF) | ((cols & 0xFFFF) << 16)),// dim1 hi | tile_dim0
                   (int)(rows & 0xFFFF),                             // tile_dim1 (tile_dim2=0)
                   (int)HW,                                          // tensor_dim0_stride lo32
                   (int)(((HW * HW) & 0xFFFF) << 16),                // dim0_str hi | dim1_str lo
                   (int)((HW * HW) >> 16) };                         // tensor_dim1_stride hi
        iv4 g2 = { 0, 0, 0, 0 };
        iv4 g3 = { 0, 0, 0, 0 };
        iv8 g4 = { 0, 0, 0, 0, 0, 0, 0, 0 };       // extra group (unused, zeros)
        __builtin_amdgcn_tensor_load_to_lds(g0, g1, g2, g3, g4, 0);
        __builtin_amdgcn_s_wait_tensorcnt(0);
    }
#else
    for (int idx = threadIdx.x; idx < rows * cols; idx += blockDim.x) {
        int lr = idx / cols, lc = idx % cols;
        raw[idx] = plane[(size_t)(r_lo + lr) * HW + (c_lo + lc)];
    }
#endif
    __syncthreads();

    // Stage 1: horizontal 7-max of thresholded values (image-OOB -> -inf)
    for (int idx = threadIdx.x; idx < rows * TILE; idx += blockDim.x) {
        int lr = idx / TILE;            // local row (relative to r_lo)
        int oc = idx % TILE;            // output col within tile
        int c = col0 + oc;
        float m = NEG_INF;
        #pragma unroll
        for (int dc = -HALO; dc <= HALO; ++dc) {
            int cc = c + dc;
            if (cc >= c_lo && cc < c_hi) {
                float v = raw[lr * cols + (cc - c_lo)];
                v = (v > med) ? v : 0.0f;
                m = fmaxf(m, v);
            }
        }
        hmax[lr * (TILE + 1) + oc] = m;
    }
    __syncthreads();

    // Stage 2: vertical 7-max + NMS compare + store
    for (int idx = threadIdx.x; idx < TILE * TILE; idx += blockDim.x) {
        int orow = idx / TILE, oc = idx % TILE;
        int r = row0 + orow;
        float b = NEG_INF;
        #pragma unroll
        for (int dr = -HALO; dr <= HALO; ++dr) {
            int rr = r + dr;
            if (rr >= r_lo && rr < r_hi)
                b = fmaxf(b, hmax[(rr - r_lo) * (TILE + 1) + oc]);
        }
        float v = raw[(r - r_lo) * cols + (col0 + oc - c_lo)];
        float xt = (v > med) ? v : 0.0f;
        out[((size_t)p * HW + r) * HW + (col0 + oc)] = (xt == b) ? xt : 0.0f;
    }
}

extern "C" void kernel_launch(void* const* d_in, const int* in_sizes, int n_in,
                              void* d_out, int out_size, void* d_ws, size_t ws_size,
                              hipStream_t stream) {
    const float* x = (const float*)d_in[0];
    float* out = (float*)d_out;
    const int n = in_sizes[0];                 // 8*1*2048*2048
    u32* ws = (u32*)d_ws;
    u32* scalars = ws;                         // [0]=hi bucket, [1]=count below, [2]=median bits
    u32* hist2   = ws + 16;                    // 65536 bins (low pass, atomics -> must be zeroed)
    u32* hist    = ws + 16 + NBINS;            // 65536 bins (high pass, written by reduce)
    u32* copies  = ws + 16 + 2 * NBINS;        // NCOPIES x 65536 per-WG histograms
    const u32 k = (u32)((n - 1) / 2);          // lower-median rank

    // zero scalars + hist2 + hist (hist needed only for fallback path; cheap)
    const int zcnt = 16 + 2 * NBINS;
    zero_kernel<<<(zcnt + 255) / 256, 256, 0, stream>>>(ws, zcnt);

    // choose number of LDS-histogram copies from available scratch
    size_t fixed_bytes = (size_t)(16 + 2 * NBINS) * 4;
    size_t avail = (ws_size > fixed_bytes) ? (ws_size - fixed_bytes) : 0;
    int ncopies = (int)(avail / ((size_t)NBINS * 4));
    if (ncopies > 256) ncopies = 256;

    if (ncopies >= 16) {
        hist_hi_lds_kernel<<<ncopies, 1024, NBINS * sizeof(u32), stream>>>(x, n, copies);
        reduce_hist_kernel<<<NBINS / 256, 256, 0, stream>>>(copies, hist, ncopies);
    } else {
        hist_hi_kernel<<<2048, 256, 0, stream>>>(x, n, hist);   // contended fallback
    }
    select_hi_kernel<<<1, 256, 0, stream>>>(hist, scalars, k);
    hist_lo_kernel<<<2048, 256, 0, stream>>>(x, n, scalars, hist2);
    select_lo_kernel<<<1, 256, 0, stream>>>(hist2, scalars, k);

    dim3 grid(HW / TILE, HW / TILE, (u32)(n / (HW * HW)));
    nms_kernel<<<grid, 256, 0, stream>>>(x, scalars, out);
}